// OptimizableSpheres_26465588478667
// MI455X (gfx1250) — compile-verified
//
#include <hip/hip_runtime.h>
#include <hip/hip_bf16.h>

typedef float v2f __attribute__((ext_vector_type(2)));
typedef float v8f __attribute__((ext_vector_type(8)));

#define N_RAYS_C   16384
#define N_SPH_C    4096
#define KSEL       256
#define RAYS_PER_BLOCK 16
#define THREADS    256

// block-wide inclusive scan of one int per thread.
// wave32: 5-step shuffle scan in-wave, then 8 wave partials combined via LDS.
// 2 barriers total (vs 17 for Hillis-Steele over LDS).
__device__ __forceinline__ int incl_scan256(int* wsum, int lane, int wave, int v) {
    int x = v;
    #pragma unroll
    for (int off = 1; off < 32; off <<= 1) {
        int y = __shfl_up(x, off, 32);
        if (lane >= off) x += y;
    }
    __syncthreads();                 // protect wsum reuse across calls
    if (lane == 31) wsum[wave] = x;
    __syncthreads();
    int base = 0;
    #pragma unroll
    for (int w = 0; w < 8; ++w) base += (w < wave) ? wsum[w] : 0;
    return x + base;
}

__global__ __launch_bounds__(THREADS) void spheres_topk_kernel(
    const float* __restrict__ rays_o, const float* __restrict__ rays_d,
    const float* __restrict__ origins, const float* __restrict__ radius_p,
    const float* __restrict__ near_p, const float* __restrict__ far_p,
    float* __restrict__ out)
{
    // LDS: 16 rays x 4096 spheres of 16-bit monotonic distance keys (128KB)
    __shared__ unsigned short keys[RAYS_PER_BLOCK][N_SPH_C];
    __shared__ int   wsum[8];
    __shared__ int   hist[256];
    __shared__ unsigned int sel[KSEL];
    __shared__ float rc[RAYS_PER_BLOCK][12]; // ox,oy,oz,dx,dy,dz,os2,od,a,near,far,pad
    __shared__ int   scal[4];                // Bhi, Clt_hi, T, Clt

    const int tid   = threadIdx.x;
    const int lane  = tid & 31;
    const int wave  = tid >> 5;
    const int half  = (lane >> 4) & 1;   // which K-pair / which C-row half
    const int l15   = lane & 15;
    const int rbase = blockIdx.x * RAYS_PER_BLOCK;

    // ---------------- ray cache ----------------
    if (tid < RAYS_PER_BLOCK) {
        int r = rbase + tid;
        float ox = rays_o[3*r+0], oy = rays_o[3*r+1], oz = rays_o[3*r+2];
        float dx = rays_d[3*r+0], dy = rays_d[3*r+1], dz = rays_d[3*r+2];
        rc[tid][0] = ox; rc[tid][1] = oy; rc[tid][2] = oz;
        rc[tid][3] = dx; rc[tid][4] = dy; rc[tid][5] = dz;
        rc[tid][6] = ox*ox + oy*oy + oz*oz;   // ||o||^2
        rc[tid][7] = ox*dx + oy*dy + oz*dz;   // o.d
        rc[tid][8] = dx*dx + dy*dy + dz*dz;   // ||d||^2 (a)
        rc[tid][9]  = near_p[r];
        rc[tid][10] = far_p[r];
    }
    __syncthreads();

    // ---------------- WMMA A operands ----------------
    // A (16x4 f32): lane row M = l15; lanes 0-15 hold K=0,1; lanes 16-31 hold K=2,3.
    // ray coord vector is (x, y, z, 0).
    v2f Ad, Ao;
    {
        float d0 = rc[l15][3], d1 = rc[l15][4], d2 = rc[l15][5];
        Ad.x = half ? d2  : d0;
        Ad.y = half ? 0.f : d1;
        float o0 = rc[l15][0], o1 = rc[l15][1], o2 = rc[l15][2];
        Ao.x = half ? o2  : o0;
        Ao.y = half ? 0.f : o1;
    }
    // per-lane epilogue ray constants: C/D row M = v + 8*half
    float os2v[8], odv[8], av[8];
    #pragma unroll
    for (int v = 0; v < 8; ++v) {
        int m = v + 8*half;
        os2v[v] = rc[m][6];
        odv[v]  = rc[m][7];
        av[v]   = rc[m][8];
    }

    // ---------------- phase 1: pairwise dist^2 via V_WMMA_F32_16X16X4_F32 ----
    // wave w owns sphere tiles [w*32, w*32+32), 16 spheres per tile.
    const int t0 = wave * 32;
    for (int t = t0; t < t0 + 32; ++t) {
        int s = t * 16 + l15;
        __builtin_prefetch(origins + 3 * (s + 16), 0, 0); // next tile hint
        float cx = origins[3*s+0], cy = origins[3*s+1], cz = origins[3*s+2];
        float cs2 = cx*cx + cy*cy + cz*cz;
        // B (4x16 f32): lane col N = l15; lanes 0-15 hold K=0,1; lanes 16-31 hold K=2,3.
        v2f B;
        B.x = half ? cz  : cx;
        B.y = half ? 0.f : cy;
        v8f zero = {};
        // G1[m][n] = d_m . c_n ; G2[m][n] = o_m . c_n
        v8f G1 = __builtin_amdgcn_wmma_f32_16x16x4_f32(false, Ad, false, B,
                                                       (short)0, zero, false, false);
        v8f G2 = __builtin_amdgcn_wmma_f32_16x16x4_f32(false, Ao, false, B,
                                                       (short)0, zero, false, false);
        #pragma unroll
        for (int v = 0; v < 8; ++v) {
            float dd = G1[v] - odv[v];                     // (c-o).d
            float n2 = cs2 - 2.f*G2[v] + os2v[v];          // ||c-o||^2
            float d2 = fmaxf(n2 * av[v] - dd * dd, 0.f);   // Lagrange identity
            keys[v + 8*half][s] = (unsigned short)(__float_as_uint(d2) >> 16);
        }
    }
    __syncthreads();

    const float radius = radius_p[0];
    const float r2 = radius * radius;
    const long long TS_OFF  = 0;
    const long long IN_OFF  = (long long)N_RAYS_C * KSEL * 2;
    const long long IDX_OFF = (long long)N_RAYS_C * KSEL * 3;

    // ---------------- phase 2+3: per-ray exact-256 radix select, sort, solve --
    for (int ry = 0; ry < RAYS_PER_BLOCK; ++ry) {
        // cache my 16 contiguous keys
        unsigned short mk[16];
        #pragma unroll
        for (int i = 0; i < 16; ++i) mk[i] = keys[ry][tid*16 + i];

        // --- level 1: histogram high byte, find bucket crossing K ---
        hist[tid] = 0;
        __syncthreads();
        #pragma unroll
        for (int i = 0; i < 16; ++i) atomicAdd(&hist[mk[i] >> 8], 1);
        __syncthreads();
        {
            int h    = hist[tid];
            int cum  = incl_scan256(wsum, lane, wave, h);
            int prev = cum - h;
            if (cum >= KSEL && prev < KSEL) { scal[0] = tid; scal[1] = prev; }
        }
        __syncthreads();
        const int Bhi = scal[0], Clt_hi = scal[1];

        // --- level 2: histogram low byte within bucket Bhi, exact threshold T ---
        hist[tid] = 0;
        __syncthreads();
        #pragma unroll
        for (int i = 0; i < 16; ++i)
            if ((mk[i] >> 8) == Bhi) atomicAdd(&hist[mk[i] & 255], 1);
        __syncthreads();
        {
            int h    = hist[tid];
            int cum  = incl_scan256(wsum, lane, wave, h) + Clt_hi;
            int prev = cum - h;
            if (cum >= KSEL && prev < KSEL) { scal[2] = (Bhi << 8) | tid; scal[3] = prev; }
        }
        __syncthreads();
        const unsigned T   = (unsigned)scal[2];
        const int      Clt = scal[3];           // #(key < T), < KSEL

        // --- deterministic index-ordered compaction into sel[0..255] ---
        int nlt = 0, neq = 0;
        #pragma unroll
        for (int i = 0; i < 16; ++i) { nlt += (mk[i] < T); neq += (mk[i] == T); }
        int own  = nlt | (neq << 16);
        int incl = incl_scan256(wsum, lane, wave, own);
        int ex   = incl - own;
        int p_lt = ex & 0xFFFF;
        int p_eq = Clt + (ex >> 16);
        #pragma unroll
        for (int i = 0; i < 16; ++i) {
            unsigned k = mk[i];
            unsigned s = (unsigned)(tid*16 + i);
            if (k < T)        sel[p_lt++] = (k << 16) | s;
            else if (k == T) { if (p_eq < KSEL) sel[p_eq] = (k << 16) | s; p_eq++; }
        }
        __syncthreads();

        // --- bitonic sort 256 packed (key<<16|idx) ascending.
        //     Element lives in a register; stride<=16 steps use in-wave
        //     lane shuffles (no barrier), only 6 cross-wave steps touch LDS. ---
        unsigned v = sel[tid];
        for (int size = 2; size <= 256; size <<= 1) {
            for (int stride = size >> 1; stride > 0; stride >>= 1) {
                unsigned p;
                if (stride >= 32) {
                    __syncthreads();
                    sel[tid] = v;
                    __syncthreads();
                    p = sel[tid ^ stride];
                } else {
                    p = (unsigned)__shfl_xor((int)v, stride, 32);
                }
                bool up    = ((tid & size)   == 0);
                bool lower = ((tid & stride) == 0);
                unsigned mn = v < p ? v : p;
                unsigned mx = v < p ? p : v;
                v = (lower == up) ? mn : mx;
            }
        }

        // --- phase 3: quadratic solve for my selected sphere, write outputs ---
        {
            int s = (int)(v & 0xFFFFu);
            int rg = rbase + ry;
            float ox = rc[ry][0], oy = rc[ry][1], oz = rc[ry][2];
            float dx = rc[ry][3], dy = rc[ry][4], dz = rc[ry][5];
            float a  = rc[ry][8], nr = rc[ry][9], fr = rc[ry][10];
            float cx = origins[3*s+0], cy = origins[3*s+1], cz = origins[3*s+2];
            float vx = ox - cx, vy = oy - cy, vz = oz - cz;     // d = o - c
            float b  = 2.f * (dx*vx + dy*vy + dz*vz);
            float c  = vx*vx + vy*vy + vz*vz - r2;
            float delta = b*b - 4.f*a*c;
            bool inter = delta > 0.f;
            float dsafe = inter ? delta : 1.f;
            float sgn = (b > 0.f) ? 1.f : ((b < 0.f) ? -1.f : 0.f);
            float q  = -0.5f * (b + sgn * sqrtf(dsafe));
            float qs = (q == 0.f) ? 1.f : q;
            float t1 = q / a, t2 = c / qs;
            float lo = fminf(t1, t2), hi = fmaxf(t1, t2);
            if (!inter) { lo = 0.f; hi = 0.f; }
            lo = fmaxf(fminf(lo, fr), nr);
            hi = fmaxf(fminf(hi, fr), nr);
            inter = inter && (hi - lo > 0.f);
            long long row = (long long)rg * KSEL + tid;
            out[TS_OFF + 2*row + 0] = lo;
            out[TS_OFF + 2*row + 1] = hi;
            out[IN_OFF  + row] = inter ? 1.f : 0.f;
            out[IDX_OFF + row] = (float)s;
        }
        __syncthreads();
    }
}

extern "C" void kernel_launch(void* const* d_in, const int* in_sizes, int n_in,
                              void* d_out, int out_size, void* d_ws, size_t ws_size,
                              hipStream_t stream) {
    const float* rays_o  = (const float*)d_in[0];
    const float* rays_d  = (const float*)d_in[1];
    const float* origins = (const float*)d_in[2];
    const float* radius  = (const float*)d_in[3];
    const float* nearp   = (const float*)d_in[4];
    const float* farp    = (const float*)d_in[5];
    float* out = (float*)d_out;
    dim3 grid(N_RAYS_C / RAYS_PER_BLOCK);  // 1024 blocks of 16 rays
    dim3 block(THREADS);
    spheres_topk_kernel<<<grid, block, 0, stream>>>(rays_o, rays_d, origins,
                                                    radius, nearp, farp, out);
}